// latentModalityLoss_72078141161682
// MI455X (gfx1250) — compile-verified
//
#include <hip/hip_runtime.h>
#include <math.h>

#define MARGIN 0.003f

typedef __attribute__((ext_vector_type(2))) float v2f;
typedef __attribute__((ext_vector_type(4))) float v4f;
typedef __attribute__((ext_vector_type(8))) float v8f;

static __device__ __forceinline__ v8f wmma_f32(v2f a, v2f b, v8f c) {
  return __builtin_amdgcn_wmma_f32_16x16x4_f32(false, a, false, b, (short)0, c,
                                               false, false);
}

// ---------------------------------------------------------------------------
// Kernel 1: split-K symmetric Gram matrix, 32x32 macro-tile per wave.
// Only the 36 upper-triangular (tm<=tn) macro-tiles are computed.
//
// Per K=16 iteration (unrolled x2): 8x global_load_b128 + 16x
// v_wmma_f32_16x16x4_f32, with the second load set folded into immediate
// offsets of the same base addresses.
// Trick: WMMA sums over whatever 4 columns the lanes supply, provided A and B
// fragments use the SAME lane->column map. Each lane loads float4 at column
// k + 4*(lane/16); .xy feeds one wmma (columns {k,k+1,k+4,k+5}) and .zw the
// next ({k+2,k+3,k+6,k+7}) — a permutation of the same K-slice.
// fp32 accumulate throughout (loss thresholds distance deltas vs 0.003).
// ---------------------------------------------------------------------------
__global__ __launch_bounds__(32) void gram_wmma_f32(
    const float* __restrict__ X, float* __restrict__ Gp,
    int B, int K, int kchunks) {
  const int NT = B / 32;               // 8 macro-tiles per side
  // map blockIdx.x in [0,36) -> (tm,tn) with tm<=tn  (uniform SALU loop)
  int rem = blockIdx.x;
  int tm = 0;
  while (rem >= NT - tm) { rem -= NT - tm; ++tm; }
  const int tn = tm + rem;

  const int kc   = blockIdx.y;
  const int lane = threadIdx.x;        // full wave, EXEC all-1s for WMMA
  const int half = lane >> 4;
  const int l16  = lane & 15;

  const int kper = K / kchunks;
  const int k0   = kc * kper;

  const float* __restrict__ a0p = X + (size_t)(tm * 32 + l16) * K + 4 * half + k0;
  const float* __restrict__ a1p = a0p + (size_t)16 * K;
  const float* __restrict__ b0p = X + (size_t)(tn * 32 + l16) * K + 4 * half + k0;
  const float* __restrict__ b1p = b0p + (size_t)16 * K;

  v8f acc00 = {}, acc01 = {}, acc10 = {}, acc11 = {};
  for (int k = 0; k < kper; k += 16) {
    // first K=8 slice
    v4f a0 = *(const v4f*)(a0p + k);
    v4f a1 = *(const v4f*)(a1p + k);
    v4f b0 = *(const v4f*)(b0p + k);
    v4f b1 = *(const v4f*)(b1p + k);
    // second K=8 slice (immediate offset from same bases)
    v4f c0 = *(const v4f*)(a0p + k + 8);
    v4f c1 = *(const v4f*)(a1p + k + 8);
    v4f d0 = *(const v4f*)(b0p + k + 8);
    v4f d1 = *(const v4f*)(b1p + k + 8);

    v2f a0lo = __builtin_shufflevector(a0, a0, 0, 1);
    v2f a0hi = __builtin_shufflevector(a0, a0, 2, 3);
    v2f a1lo = __builtin_shufflevector(a1, a1, 0, 1);
    v2f a1hi = __builtin_shufflevector(a1, a1, 2, 3);
    v2f b0lo = __builtin_shufflevector(b0, b0, 0, 1);
    v2f b0hi = __builtin_shufflevector(b0, b0, 2, 3);
    v2f b1lo = __builtin_shufflevector(b1, b1, 0, 1);
    v2f b1hi = __builtin_shufflevector(b1, b1, 2, 3);

    acc00 = wmma_f32(a0lo, b0lo, acc00);
    acc00 = wmma_f32(a0hi, b0hi, acc00);
    acc01 = wmma_f32(a0lo, b1lo, acc01);
    acc01 = wmma_f32(a0hi, b1hi, acc01);
    acc10 = wmma_f32(a1lo, b0lo, acc10);
    acc10 = wmma_f32(a1hi, b0hi, acc10);
    acc11 = wmma_f32(a1lo, b1lo, acc11);
    acc11 = wmma_f32(a1hi, b1hi, acc11);

    v2f c0lo = __builtin_shufflevector(c0, c0, 0, 1);
    v2f c0hi = __builtin_shufflevector(c0, c0, 2, 3);
    v2f c1lo = __builtin_shufflevector(c1, c1, 0, 1);
    v2f c1hi = __builtin_shufflevector(c1, c1, 2, 3);
    v2f d0lo = __builtin_shufflevector(d0, d0, 0, 1);
    v2f d0hi = __builtin_shufflevector(d0, d0, 2, 3);
    v2f d1lo = __builtin_shufflevector(d1, d1, 0, 1);
    v2f d1hi = __builtin_shufflevector(d1, d1, 2, 3);

    acc00 = wmma_f32(c0lo, d0lo, acc00);
    acc00 = wmma_f32(c0hi, d0hi, acc00);
    acc01 = wmma_f32(c0lo, d1lo, acc01);
    acc01 = wmma_f32(c0hi, d1hi, acc01);
    acc10 = wmma_f32(c1lo, d0lo, acc10);
    acc10 = wmma_f32(c1hi, d0hi, acc10);
    acc11 = wmma_f32(c1lo, d1lo, acc11);
    acc11 = wmma_f32(c1hi, d1hi, acc11);
  }

  // C/D layout: VGPR r -> (lanes 0-15: M=r, N=lane), (lanes 16-31: M=r+8)
  float* __restrict__ Gout = Gp + (size_t)kc * B * B;
  const int n0 = tn * 32 + l16;
  const int n1 = n0 + 16;
  const int m0 = tm * 32 + half * 8;
  const int m1 = m0 + 16;
#pragma unroll
  for (int r = 0; r < 8; ++r) {
    Gout[(size_t)(m0 + r) * B + n0] = acc00[r];
    Gout[(size_t)(m0 + r) * B + n1] = acc01[r];
    Gout[(size_t)(m1 + r) * B + n0] = acc10[r];
    Gout[(size_t)(m1 + r) * B + n1] = acc11[r];
  }
}

// ---------------------------------------------------------------------------
// Kernel 2: deterministically sum K-chunk partials (upper-triangular storage)
// and form D[a,b] = sqrt(max(G[a,a]+G[b,b]-2G[a,b], 0))
// ---------------------------------------------------------------------------
__global__ void dist_kernel(const float* __restrict__ Gp, float* __restrict__ Dm,
                            int B, int kchunks) {
  int idx = blockIdx.x * blockDim.x + threadIdx.x;
  if (idx >= B * B) return;
  int a = idx / B;
  int b = idx - a * B;
  int lo = a < b ? a : b;
  int hi = a < b ? b : a;
  float gab = 0.f, gaa = 0.f, gbb = 0.f;
  for (int c = 0; c < kchunks; ++c) {
    const float* Gc = Gp + (size_t)c * B * B;
    gab += Gc[(size_t)lo * B + hi];
    gaa += Gc[(size_t)a * B + a];
    gbb += Gc[(size_t)b * B + b];
  }
  float d2 = gaa + gbb - 2.f * gab;
  d2 = fmaxf(d2, 0.f);
  Dm[idx] = (d2 > 0.f) ? sqrtf(d2) : 0.f;
}

// ---------------------------------------------------------------------------
// Kernel 3: per-(i,j) anchor-positive pair, reduce over impostors k.
// One wave32 per pair; deterministic partial sums (no float atomics).
// ---------------------------------------------------------------------------
__global__ __launch_bounds__(32) void triplet_kernel(
    const float* __restrict__ Dm, const int* __restrict__ mods,
    float* __restrict__ partial, int B) {
  const int pair = blockIdx.x;
  const int i = pair / B;
  const int j = pair - i * B;
  const int lane = threadIdx.x;

  float s = 0.f, cnt = 0.f;
  const bool valid = (i < j) && (mods[i] == mods[j]);
  if (valid) {
    const float dij = Dm[(size_t)i * B + j];
    const int mi = mods[i];
    for (int k = lane; k < B; k += 32) {
      if (mods[k] != mi) {
        float t1 = fmaxf(dij - Dm[(size_t)i * B + k] + MARGIN, 0.f);
        float t2 = fmaxf(dij - Dm[(size_t)j * B + k] + MARGIN, 0.f);
        s += t1 + t2;
        cnt += 1.f;
      }
    }
  }
#pragma unroll
  for (int off = 16; off > 0; off >>= 1) {
    s   += __shfl_down(s, off, 32);
    cnt += __shfl_down(cnt, off, 32);
  }
  if (lane == 0) {
    partial[pair] = s;
    partial[B * B + pair] = cnt;
  }
}

// ---------------------------------------------------------------------------
// Kernel 4: single-block deterministic tree reduction -> loss scalar
// loss = sum_terms / (2 * sum_mask)
// ---------------------------------------------------------------------------
__global__ void finalize_kernel(const float* __restrict__ partial,
                                float* __restrict__ out, int N) {
  __shared__ float ss[256];
  __shared__ float sc[256];
  float s = 0.f, c = 0.f;
  for (int t = threadIdx.x; t < N; t += 256) {
    s += partial[t];
    c += partial[N + t];
  }
  ss[threadIdx.x] = s;
  sc[threadIdx.x] = c;
  __syncthreads();
  for (int off = 128; off > 0; off >>= 1) {
    if ((int)threadIdx.x < off) {
      ss[threadIdx.x] += ss[threadIdx.x + off];
      sc[threadIdx.x] += sc[threadIdx.x + off];
    }
    __syncthreads();
  }
  if (threadIdx.x == 0) {
    out[0] = ss[0] / (2.f * sc[0]);
  }
}

extern "C" void kernel_launch(void* const* d_in, const int* in_sizes, int n_in,
                              void* d_out, int out_size, void* d_ws, size_t ws_size,
                              hipStream_t stream) {
  const float* X    = (const float*)d_in[0];   // [256, 32768] fp32
  const int*   mods = (const int*)d_in[1];     // [256] int32
  float* out = (float*)d_out;                  // scalar fp32

  const int B = 256;
  const int K = in_sizes[0] / B;               // 32768
  const int KCHUNKS = 32;
  const int NT = B / 32;                       // 8
  const int NTILES = NT * (NT + 1) / 2;        // 36 upper-triangular tiles

  float* Gp      = (float*)d_ws;               // [KCHUNKS, B, B]
  float* Dm      = Gp + (size_t)KCHUNKS * B * B;
  float* partial = Dm + (size_t)B * B;         // [2, B*B]

  dim3 gridG(NTILES, KCHUNKS);                 // 1152 single-wave workgroups
  gram_wmma_f32<<<gridG, 32, 0, stream>>>(X, Gp, B, K, KCHUNKS);

  dist_kernel<<<(B * B + 255) / 256, 256, 0, stream>>>(Gp, Dm, B, KCHUNKS);

  triplet_kernel<<<B * B, 32, 0, stream>>>(Dm, mods, partial, B);

  finalize_kernel<<<1, 256, 0, stream>>>(partial, out, B * B);
}